// EdgeToAtomLayer_78082505441594
// MI455X (gfx1250) — compile-verified
//
#include <hip/hip_runtime.h>

// Scatter-add (jax.ops.segment_sum) for edge_attr[E,128] -> out[N,128] by
// edge_index[1,:]. Memory/atomic bound: ~260MB streamed reads + 64M f32
// atomics into a 25.6MB output region that stays resident in MI455X's 192MB
// L2. Non-temporal loads on the streamed inputs protect that residency.

typedef __attribute__((ext_vector_type(4))) float v4f;

#define F_DIM 128   // feature dim (floats per edge) = 32 lanes * float4

// ---------------------------------------------------------------------------
// Kernel: one wave32 per edge. Lane l loads float4 at features [4l..4l+3]
// (one coalesced 512B b128 sweep per edge, non-temporal), then issues four
// no-return global_atomic_add_f32 into out[dst, :]. The atomics resolve in
// L2 (output region fits trivially in 192MB).
// Defined first so its assembly leads the disasm snippet.
// ---------------------------------------------------------------------------
__global__ __launch_bounds__(256) void scatter_add_kernel(
    const float* __restrict__ edge_attr,   // [E, 128]
    const int*   __restrict__ edge_index,  // [2, E] row-major; row 1 = targets
    float*       __restrict__ out,         // [N, 128]
    int E) {
  const int gtid = blockIdx.x * blockDim.x + threadIdx.x;
  const int edge = gtid >> 5;          // wave32: one edge per wave
  const int lane = gtid & 31;
  if (edge >= E) return;

  // Target node for this edge (row 1 of edge_index). Uniform across the
  // wave; all 32 lane requests coalesce to a single L2 access. Read-once
  // stream -> non-temporal.
  const int dst = __builtin_nontemporal_load(edge_index + E + edge);

  // Non-temporal 128-bit load of this lane's 4 features: edge_attr is
  // read exactly once (256MB stream) -> TH=NT so it does not evict the
  // L2-resident output accumulation region.
  const v4f* src =
      reinterpret_cast<const v4f*>(edge_attr) + (size_t)edge * 32 + lane;
  const v4f v = __builtin_nontemporal_load(src);

  float* o = out + (size_t)dst * F_DIM + (lane << 2);
  // Result discarded -> non-returning global_atomic_add_f32 (STOREcnt).
  (void)__hip_atomic_fetch_add(o + 0, v.x, __ATOMIC_RELAXED,
                               __HIP_MEMORY_SCOPE_AGENT);
  (void)__hip_atomic_fetch_add(o + 1, v.y, __ATOMIC_RELAXED,
                               __HIP_MEMORY_SCOPE_AGENT);
  (void)__hip_atomic_fetch_add(o + 2, v.z, __ATOMIC_RELAXED,
                               __HIP_MEMORY_SCOPE_AGENT);
  (void)__hip_atomic_fetch_add(o + 3, v.w, __ATOMIC_RELAXED,
                               __HIP_MEMORY_SCOPE_AGENT);
}

// ---------------------------------------------------------------------------
// Kernel: zero the output (harness poisons d_out; also pre-warms L2 with
// the accumulation region right before the atomic phase).
// ---------------------------------------------------------------------------
__global__ __launch_bounds__(256) void zero_out_kernel(v4f* __restrict__ out,
                                                       int n_vec4) {
  int i = blockIdx.x * blockDim.x + threadIdx.x;
  if (i < n_vec4) {
    v4f z = {0.0f, 0.0f, 0.0f, 0.0f};
    out[i] = z;   // regular-temporal store: keep lines dirty-resident in L2
  }
}

// ---------------------------------------------------------------------------
// Launch
// ---------------------------------------------------------------------------
extern "C" void kernel_launch(void* const* d_in, const int* in_sizes, int n_in,
                              void* d_out, int out_size, void* d_ws,
                              size_t ws_size, hipStream_t stream) {
  const float* edge_attr  = (const float*)d_in[0];  // [E, 128] f32
  const int*   edge_index = (const int*)d_in[1];    // [2, E] int
  float*       out        = (float*)d_out;          // [N, 128] f32

  const int E = in_sizes[0] / F_DIM;

  // Phase 1: zero output (vectorized b128 stores).
  const int n_vec4 = out_size / 4;
  {
    dim3 block(256);
    dim3 grid((n_vec4 + 255) / 256);
    zero_out_kernel<<<grid, block, 0, stream>>>((v4f*)out, n_vec4);
  }

  // Phase 2: wave-per-edge scatter-add. 32 threads per edge.
  {
    const long long total_threads = (long long)E * 32;
    dim3 block(256);
    dim3 grid((unsigned int)((total_threads + 255) / 256));
    scatter_add_kernel<<<grid, block, 0, stream>>>(edge_attr, edge_index, out,
                                                   E);
  }
}